// NodeModel_1589137900000
// MI455X (gfx1250) — compile-verified
//
#include <hip/hip_runtime.h>

typedef __attribute__((ext_vector_type(2))) float v2f;
typedef __attribute__((ext_vector_type(8))) float v8f;

#define HDIM 128
#define TWOH 256
#define LDSP 260   // padded pitch: 260 % 64 == 4 -> conflict-free column walks

// ---------------------------------------------------------------------------
// Kernel 1: scatter-accumulate edge_attr rows into summed[col], count edges.
// One wave32 per edge: 32 lanes x 4 floats = 128-wide row, coalesced atomics.
// ---------------------------------------------------------------------------
__global__ __launch_bounds__(256) void scatter_accum_kernel(
    const float* __restrict__ edge_attr,
    const long long* __restrict__ edge_index,   // [2, n_edges], row 1 = col
    float* __restrict__ summed,                 // [n_nodes, 128] (pre-zeroed)
    float* __restrict__ cnt,                    // [n_nodes]      (pre-zeroed)
    int n_edges)
{
    const int gid  = blockIdx.x * blockDim.x + threadIdx.x;
    const int e    = gid >> 5;
    const int lane = threadIdx.x & 31;
    if (e >= n_edges) return;

    const long long col = edge_index[(long long)n_edges + e];
    const float* __restrict__ src = edge_attr + (long long)e * HDIM;
    float* __restrict__ dst = summed + col * HDIM;

#pragma unroll
    for (int i = 0; i < HDIM / 32; ++i)
        atomicAdd(&dst[lane + 32 * i], src[lane + 32 * i]);
    if (lane == 0)
        atomicAdd(&cnt[col], 1.0f);
}

// ---------------------------------------------------------------------------
// Kernel 2: fused  concat -> GEMM1(256x256) -> LayerNorm -> PReLU ->
//                  GEMM2(256x128) -> out
// 16-node tile per block, 512 threads = 16 waves (wave32).
// GEMMs use v_wmma_f32_16x16x4_f32 (native fp32 matrix path).
// ---------------------------------------------------------------------------
__global__ __launch_bounds__(512) void node_mlp_fused_kernel(
    const float* __restrict__ x,
    const float* __restrict__ summed,
    const float* __restrict__ cnt,
    const float* __restrict__ W1,      // [256,256] row-major
    const float* __restrict__ b1,      // [256]
    const float* __restrict__ gamma,   // [256]
    const float* __restrict__ beta,    // [256]
    const float* __restrict__ prelu_a, // [1]
    const float* __restrict__ W2,      // [256,128] row-major
    const float* __restrict__ b2,      // [128]
    float* __restrict__ out)           // [n_nodes,128]
{
    __shared__ float sA[16][LDSP];   // concat(x, agg)  : 16 x 256
    __shared__ float sH[16][LDSP];   // hidden after G1 : 16 x 256

    const int t    = threadIdx.x;
    const int lane = t & 31;
    const int wv   = t >> 5;          // wave id 0..15
    const int hi   = lane >> 4;       // lane half (0: lanes 0-15, 1: 16-31)
    const int l16  = lane & 15;
    const long long node0 = (long long)blockIdx.x * 16;

    // ---- Stage concat(x, summed/max(cnt,1)) into LDS: 4096 elems, 8/thread
#pragma unroll
    for (int i = 0; i < 8; ++i) {
        const int idx = t + 512 * i;
        const int row = idx >> 8;         // 0..15
        const int col = idx & 255;        // 0..255
        const long long node = node0 + row;
        float v;
        if (col < HDIM) {
            v = x[node * HDIM + col];
        } else {
            const float c   = cnt[node];
            const float inv = 1.0f / fmaxf(c, 1.0f);
            v = summed[node * HDIM + (col - HDIM)] * inv;
        }
        sA[row][col] = v;
    }
    __syncthreads();

    // ---- GEMM1: (16x256) @ W1(256x256); wave wv owns N-tile [wv*16, wv*16+16)
    {
        const int n0 = wv * 16;
        v8f acc = {};
#pragma unroll 4
        for (int k0 = 0; k0 < TWOH; k0 += 4) {
            const int ka = k0 + 2 * hi;   // this lane-half's K pair
            v2f a, b;
            // A fragment (16x4, MxK): lanes 0-15 -> K=k0,k0+1 ; 16-31 -> k0+2,k0+3
            a.x = sA[l16][ka];
            a.y = sA[l16][ka + 1];
            // B fragment (4x16, KxN), W1 row-major
            b.x = W1[(ka)     * TWOH + n0 + l16];
            b.y = W1[(ka + 1) * TWOH + n0 + l16];
            acc = __builtin_amdgcn_wmma_f32_16x16x4_f32(
                /*neg_a=*/false, a, /*neg_b=*/false, b,
                /*c_mod=*/(short)0, acc, /*reuse_a=*/false, /*reuse_b=*/false);
        }
        // C/D layout: VGPR r -> M = r + 8*hi, N = n0 + l16
#pragma unroll
        for (int r = 0; r < 8; ++r)
            sH[r + 8 * hi][n0 + l16] = acc[r] + b1[n0 + l16];
    }
    __syncthreads();

    // ---- LayerNorm (eps 1e-5) + PReLU: wave wv owns row wv, 8 cols per lane
    {
        float v[8];
        float s1 = 0.f, s2 = 0.f;
#pragma unroll
        for (int i = 0; i < 8; ++i) {
            v[i] = sH[wv][lane + 32 * i];
            s1 += v[i];
            s2 += v[i] * v[i];
        }
#pragma unroll
        for (int m = 16; m >= 1; m >>= 1) {
            s1 += __shfl_xor(s1, m, 32);
            s2 += __shfl_xor(s2, m, 32);
        }
        const float mu   = s1 * (1.0f / 256.0f);
        const float var  = s2 * (1.0f / 256.0f) - mu * mu;
        const float rsig = rsqrtf(var + 1e-5f);
        const float pa   = prelu_a[0];
#pragma unroll
        for (int i = 0; i < 8; ++i) {
            const int c = lane + 32 * i;
            float h = (v[i] - mu) * rsig * gamma[c] + beta[c];
            sH[wv][c] = (h >= 0.f) ? h : pa * h;
        }
    }
    __syncthreads();

    // ---- GEMM2: (16x256) @ W2(256x128); waves 0..7 own the 8 N-tiles
    if (wv < 8) {
        const int n0 = wv * 16;
        v8f acc = {};
#pragma unroll 4
        for (int k0 = 0; k0 < TWOH; k0 += 4) {
            const int ka = k0 + 2 * hi;
            v2f a, b;
            a.x = sH[l16][ka];
            a.y = sH[l16][ka + 1];
            b.x = W2[(ka)     * HDIM + n0 + l16];
            b.y = W2[(ka + 1) * HDIM + n0 + l16];
            acc = __builtin_amdgcn_wmma_f32_16x16x4_f32(
                false, a, false, b, (short)0, acc, false, false);
        }
#pragma unroll
        for (int r = 0; r < 8; ++r) {
            const int row = r + 8 * hi;
            out[(node0 + row) * HDIM + n0 + l16] = acc[r] + b2[n0 + l16];
        }
    }
}

// ---------------------------------------------------------------------------
extern "C" void kernel_launch(void* const* d_in, const int* in_sizes, int n_in,
                              void* d_out, int out_size, void* d_ws, size_t ws_size,
                              hipStream_t stream)
{
    const float*     x          = (const float*)d_in[0];
    const long long* edge_index = (const long long*)d_in[1];   // int64 in reference
    const float*     edge_attr  = (const float*)d_in[2];
    const float*     W1         = (const float*)d_in[3];
    const float*     b1         = (const float*)d_in[4];
    const float*     gamma      = (const float*)d_in[5];
    const float*     beta       = (const float*)d_in[6];
    const float*     prelu_a    = (const float*)d_in[7];
    const float*     W2         = (const float*)d_in[8];
    const float*     b2         = (const float*)d_in[9];
    float*           out        = (float*)d_out;

    const int n_nodes = in_sizes[0] / HDIM;       // 100000
    const int n_edges = in_sizes[1] / 2;          // 600000

    float* summed = (float*)d_ws;                              // [n_nodes,128]
    float* cnt    = summed + (size_t)n_nodes * HDIM;           // [n_nodes]

    // Zero the scatter workspace every call (graph-capture safe, deterministic).
    const size_t clear_bytes = ((size_t)n_nodes * HDIM + (size_t)n_nodes) * sizeof(float);
    hipMemsetAsync(d_ws, 0, clear_bytes, stream);

    // Scatter: one wave per edge, 8 waves (256 threads) per block.
    const int sblocks = (n_edges + 7) / 8;
    scatter_accum_kernel<<<sblocks, 256, 0, stream>>>(
        edge_attr, edge_index, summed, cnt, n_edges);

    // Fused MLP: 16 nodes per block, 512 threads (16 waves).
    const int mblocks = (n_nodes + 15) / 16;
    node_mlp_fused_kernel<<<mblocks, 512, 0, stream>>>(
        x, summed, cnt, W1, b1, gamma, beta, prelu_a, W2, b2, out);
}